// SkipGramModel_63290638073916
// MI455X (gfx1250) — compile-verified
//
#include <hip/hip_runtime.h>
#include <hip/hip_bf16.h>

// CDNA5 / gfx1250 skip-gram negative-sampling loss.
// Strategy: the gather tables (51.2 MB each) are L2-resident on MI455X
// (192 MB L2); the kernel is bound by ~600 MB of L2 gather traffic, so
// compute is free. We use V_WMMA_F32_16X16X4_F32 to compute 16x16 blocks
// of (label-row x center-row) dot products for tiles of 16 batch elements,
// consuming only the diagonal. f32 WMMA keeps reference precision.

typedef __attribute__((ext_vector_type(2))) float v2f;
typedef __attribute__((ext_vector_type(8))) float v8f;

#define DIM   128
#define WPOS  10
#define KNEG  64
#define NLAB  (WPOS + KNEG)   // 74
#define BTILE 16
#define WAVES_PER_BLOCK 4

__device__ __forceinline__ float log_sigmoid(float x) {
    // Stable: x>=0 -> -log1p(exp(-x)); x<0 -> x - log1p(exp(x))
    return fminf(x, 0.0f) - log1pf(__expf(-fabsf(x)));
}

__global__ __launch_bounds__(128) void sgns_wmma_kernel(
    const int*   __restrict__ input,       // [B]
    const int*   __restrict__ pos_labels,  // [B, WPOS]
    const int*   __restrict__ neg_labels,  // [B, KNEG]
    const float* __restrict__ input_emb,   // [VOCAB, DIM]
    const float* __restrict__ output_emb,  // [VOCAB, DIM]
    float*       __restrict__ partials)    // [gridDim.x]
{
    const int lane = threadIdx.x & 31;
    const int wave = threadIdx.x >> 5;
    const int m    = lane & 15;      // batch row within tile (A row / B col)
    const int h    = lane >> 4;      // K-half selector within a 4-wide chunk
    const int b    = blockIdx.x * BTILE + m;

    // ---- Preload B-matrix (centers) chunks into registers, WMMA B layout.
    // B[k][n] = C[n][4c+k]; lane l supplies column n=l&15, K = 2h, 2h+1.
    const long cidx = input[b];
    const float* crow = input_emb + cidx * (long)DIM;
    v2f Bc[32];
#pragma unroll
    for (int c = 0; c < 32; ++c)
        Bc[c] = *(const v2f*)(crow + c * 4 + h * 2);

    // ---- Label range for this wave: 74 labels split 19/19/18/18.
    const int jbeg = wave * 18 + (wave < 2 ? wave : 2);
    const int jend = jbeg + ((wave < 2) ? 19 : 18);

    // Diagonal element (m,m) of the 16x16 f32 D tile lives in:
    //   m in [0,8):  lane m,     VGPR m
    //   m in [8,16): lane m+16,  VGPR m-8
    const bool diag_lane = (lane < 8) || (lane >= 24);
    const int  r = lane & 7;

    float lane_sum = 0.0f;

    for (int j = jbeg; j < jend; ++j) {
        const long lbl = (j < WPOS) ? pos_labels[b * WPOS + j]
                                    : neg_labels[b * KNEG + (j - WPOS)];
        const float* arow = output_emb + lbl * (long)DIM;

        // Prefetch next label row into cache (global_prefetch_b8).
        if (j + 1 < jend) {
            const long lbl2 = (j + 1 < WPOS) ? pos_labels[b * WPOS + j + 1]
                                             : neg_labels[b * KNEG + (j + 1 - WPOS)];
            __builtin_prefetch(output_emb + lbl2 * (long)DIM + h * 64, 0, 0);
        }

        // A-matrix: row m = label row of batch b; lane loads K = 4c+2h, +1.
        v8f acc = {0.f, 0.f, 0.f, 0.f, 0.f, 0.f, 0.f, 0.f};
#pragma unroll
        for (int c = 0; c < 32; ++c) {
            v2f a = *(const v2f*)(arow + c * 4 + h * 2);
            acc = __builtin_amdgcn_wmma_f32_16x16x4_f32(
                /*neg_a=*/false, a, /*neg_b=*/false, Bc[c],
                /*c_mod=*/(short)0, acc, /*reuse_a=*/false, /*reuse_b=*/false);
        }

        // Extract this lane's diagonal element (branchless cndmask chain).
        float v = acc[0];
        v = (r == 1) ? acc[1] : v;
        v = (r == 2) ? acc[2] : v;
        v = (r == 3) ? acc[3] : v;
        v = (r == 4) ? acc[4] : v;
        v = (r == 5) ? acc[5] : v;
        v = (r == 6) ? acc[6] : v;
        v = (r == 7) ? acc[7] : v;

        lane_sum += diag_lane ? log_sigmoid(v) : 0.0f;
    }

    // ---- Deterministic wave reduction (fixed shuffle tree, wave32).
#pragma unroll
    for (int off = 16; off > 0; off >>= 1)
        lane_sum += __shfl_down(lane_sum, off, 32);

    __shared__ float wsum[WAVES_PER_BLOCK];
    if (lane == 0) wsum[wave] = lane_sum;
    __syncthreads();
    if (threadIdx.x == 0) {
        float s = 0.0f;
#pragma unroll
        for (int w = 0; w < WAVES_PER_BLOCK; ++w) s += wsum[w];
        partials[blockIdx.x] = s;   // plain store: no float atomics -> deterministic
    }
}

__global__ __launch_bounds__(256) void sgns_reduce_kernel(
    const float* __restrict__ partials, float* __restrict__ out, int n)
{
    __shared__ float red[256];
    const int t = threadIdx.x;
    float s = 0.0f;
    for (int i = t; i < n; i += 256) s += partials[i];  // fixed order
    red[t] = s;
    __syncthreads();
#pragma unroll
    for (int off = 128; off > 0; off >>= 1) {
        if (t < off) red[t] += red[t + off];
        __syncthreads();
    }
    if (t == 0) out[0] = -red[0];   // loss = -(sum of all logsigmoids)
}

extern "C" void kernel_launch(void* const* d_in, const int* in_sizes, int n_in,
                              void* d_out, int out_size, void* d_ws, size_t ws_size,
                              hipStream_t stream)
{
    const int*   input      = (const int*)  d_in[0];  // [B]
    const int*   pos_labels = (const int*)  d_in[1];  // [B,10]
    const int*   neg_labels = (const int*)  d_in[2];  // [B,64]
    const float* input_emb  = (const float*)d_in[3];  // [VOCAB,128]
    const float* output_emb = (const float*)d_in[4];  // [VOCAB,128]
    float*       out        = (float*)d_out;
    float*       partials   = (float*)d_ws;           // nblocks floats

    const int B = in_sizes[0];          // 16384
    const int nblocks = B / BTILE;      // 1024

    sgns_wmma_kernel<<<nblocks, 32 * WAVES_PER_BLOCK, 0, stream>>>(
        input, pos_labels, neg_labels, input_emb, output_emb, partials);
    sgns_reduce_kernel<<<1, 256, 0, stream>>>(partials, out, nblocks);
}